// Model_87058987090632
// MI455X (gfx1250) — compile-verified
//
#include <hip/hip_runtime.h>
#include <math.h>

#define BATCHN 128
#define SEQ    2048
#define ENC    8
#define DMODEL 16
#define DINNER 32
#define DSTATE 16
#define DCONV  4
#define PRED   720
#define NTOK   (BATCHN * SEQ)
#define CHUNK  256
#define NCHUNK (SEQ / CHUNK)

typedef __attribute__((ext_vector_type(2))) float v2f;
typedef __attribute__((ext_vector_type(8))) float v8f;

__device__ __forceinline__ float silu_f(float v) { return v / (1.0f + __expf(-v)); }
__device__ __forceinline__ float softplus_f(float v) { return v > 20.0f ? v : log1pf(__expf(v)); }

__device__ __forceinline__ void layernorm16(const float* __restrict__ v,
                                            const float* __restrict__ g,
                                            const float* __restrict__ b,
                                            float* __restrict__ o) {
    float m = 0.0f;
#pragma unroll
    for (int i = 0; i < 16; ++i) m += v[i];
    m *= (1.0f / 16.0f);
    float var = 0.0f;
#pragma unroll
    for (int i = 0; i < 16; ++i) { float d = v[i] - m; var += d * d; }
    var *= (1.0f / 16.0f);
    float r = rsqrtf(var + 1e-5f);
#pragma unroll
    for (int i = 0; i < 16; ++i) o[i] = (v[i] - m) * r * g[i] + b[i];
}

// ---------------------------------------------------------------------------
// Kernel 1: x -> silu(proj1) -> proj2 -> relu(LN) = residual -> LN -> in_proj
// ---------------------------------------------------------------------------
__global__ __launch_bounds__(256) void front_kernel(
    const float* __restrict__ x,
    const float* __restrict__ w1, const float* __restrict__ b1,
    const float* __restrict__ w2, const float* __restrict__ b2,
    const float* __restrict__ lng, const float* __restrict__ lnb,
    const float* __restrict__ bng, const float* __restrict__ bnb,
    const float* __restrict__ win,
    float* __restrict__ res, float* __restrict__ xpre, float* __restrict__ zbuf)
{
    int t = blockIdx.x * 256 + threadIdx.x;
    if (t >= NTOK) return;

    float xin[ENC];
#pragma unroll
    for (int i = 0; i < ENC; ++i) xin[i] = x[(size_t)t * ENC + i];

    float h1[8];
#pragma unroll
    for (int o = 0; o < 8; ++o) {
        float s = b1[o];
#pragma unroll
        for (int i = 0; i < 8; ++i) s += w1[o * 8 + i] * xin[i];
        h1[o] = silu_f(s);
    }

    float h2[16];
#pragma unroll
    for (int o = 0; o < 16; ++o) {
        float s = b2[o];
#pragma unroll
        for (int i = 0; i < 8; ++i) s += w2[o * 8 + i] * h1[i];
        h2[o] = s;
    }

    float r[16];
    layernorm16(h2, lng, lnb, r);
#pragma unroll
    for (int i = 0; i < 16; ++i) r[i] = fmaxf(r[i], 0.0f);
#pragma unroll
    for (int i = 0; i < 16; ++i) res[(size_t)t * 16 + i] = r[i];

    float h3[16];
    layernorm16(r, bng, bnb, h3);

    for (int o = 0; o < DINNER; ++o) {
        float sx = 0.0f, sz = 0.0f;
#pragma unroll
        for (int i = 0; i < 16; ++i) {
            sx += win[o * 16 + i] * h3[i];
            sz += win[(DINNER + o) * 16 + i] * h3[i];
        }
        xpre[(size_t)t * DINNER + o] = sx;
        zbuf[(size_t)t * DINNER + o] = sz;
    }
}

// ---------------------------------------------------------------------------
// Kernel 2: causal depthwise conv + silu, then x_proj (33x32) + softplus(dt)
// ---------------------------------------------------------------------------
__global__ __launch_bounds__(256) void conv_proj_kernel(
    const float* __restrict__ xpre,
    const float* __restrict__ convw, const float* __restrict__ convb,
    const float* __restrict__ xprojw,
    const float* __restrict__ dtw, const float* __restrict__ dtb,
    float* __restrict__ xc_out, float* __restrict__ dt_out,
    float* __restrict__ B_out, float* __restrict__ C_out)
{
    int t = blockIdx.x * 256 + threadIdx.x;
    if (t >= NTOK) return;
    int l = t & (SEQ - 1);

    float xc[DINNER];
    for (int d = 0; d < DINNER; ++d) {
        float s = convb[d];
#pragma unroll
        for (int k = 0; k < DCONV; ++k) {
            int ls = l - (DCONV - 1) + k;
            if (ls >= 0)
                s += convw[d * DCONV + k] * xpre[(size_t)(t - (DCONV - 1) + k) * DINNER + d];
        }
        xc[d] = silu_f(s);
        xc_out[(size_t)t * DINNER + d] = xc[d];
    }

    float dtr = 0.0f;
#pragma unroll
    for (int d = 0; d < DINNER; ++d) dtr += xprojw[d] * xc[d];

    for (int j = 0; j < DSTATE; ++j) {
        float sb = 0.0f, sc = 0.0f;
#pragma unroll
        for (int d = 0; d < DINNER; ++d) {
            sb += xprojw[(1 + j) * DINNER + d] * xc[d];
            sc += xprojw[(1 + DSTATE + j) * DINNER + d] * xc[d];
        }
        B_out[(size_t)t * DSTATE + j] = sb;
        C_out[(size_t)t * DSTATE + j] = sc;
    }

    for (int d = 0; d < DINNER; ++d)
        dt_out[(size_t)t * DINNER + d] = softplus_f(dtr * dtw[d] + dtb[d]);
}

// ---------------------------------------------------------------------------
// Chunked selective scan (3 phases). Linear recurrence h_l = dA_l*h_{l-1}+u_l
// is split into NCHUNK chunks of CHUNK steps to cut the dependent-chain
// latency ~4x. Lane map (per wave): lanes 0-15 -> (d0,n), 16-31 -> (d0+1,n).
// ---------------------------------------------------------------------------

// Phase A: local scan per chunk with h0 = 0; emit end-state and prod(dA).
__global__ __launch_bounds__(256) void scan_local_kernel(
    const float* __restrict__ dt, const float* __restrict__ xc,
    const float* __restrict__ Bm, const float* __restrict__ A_log,
    float* __restrict__ hend, float* __restrict__ pprod)
{
    int lane = threadIdx.x & 31;
    int wave = threadIdx.x >> 5;
    int c  = blockIdx.x;                 // chunk index
    int bh = blockIdx.y;                 // b*2 + half
    int b  = bh >> 1;
    int d  = (bh & 1) * 16 + wave * 2 + (lane >> 4);
    int n  = lane & 15;

    float a = -__expf(A_log[d * DSTATE + n]);
    size_t base = (size_t)b * SEQ + (size_t)c * CHUNK;
    const float* dtp = dt + base * DINNER + d;
    const float* xcp = xc + base * DINNER + d;
    const float* Bp  = Bm + base * DSTATE + n;

    float h = 0.0f, p = 1.0f;
    for (int l = 0; l < CHUNK; ++l) {
        float dtv = dtp[(size_t)l * DINNER];
        float xv  = xcp[(size_t)l * DINNER];
        float bv  = Bp[(size_t)l * DSTATE];
        float dA  = __expf(dtv * a);
        h = fmaf(dA, h, dtv * bv * xv);
        p *= dA;
    }
    size_t si = ((((size_t)b * DINNER) + d) * DSTATE + n) * NCHUNK + c;
    hend[si]  = h;
    pprod[si] = p;
}

// Phase B: sequence the NCHUNK chunk summaries -> true initial state per chunk.
__global__ __launch_bounds__(256) void scan_combine_kernel(
    const float* __restrict__ hend, const float* __restrict__ pprod,
    float* __restrict__ hinit)
{
    int e = blockIdx.x * 256 + threadIdx.x;   // over B*DINNER*DSTATE = 65536
    if (e >= BATCHN * DINNER * DSTATE) return;
    size_t base = (size_t)e * NCHUNK;
    float H = 0.0f;
    for (int c = 0; c < NCHUNK; ++c) {
        hinit[base + c] = H;
        H = fmaf(pprod[base + c], H, hend[base + c]);
    }
}

// Phase C: re-scan each chunk from its true initial state; emit y.
__global__ __launch_bounds__(256) void scan_final_kernel(
    const float* __restrict__ dt, const float* __restrict__ xc,
    const float* __restrict__ Bm, const float* __restrict__ Cm,
    const float* __restrict__ A_log, const float* __restrict__ hinit,
    float* __restrict__ ybuf)
{
    int lane = threadIdx.x & 31;
    int wave = threadIdx.x >> 5;
    int c  = blockIdx.x;
    int bh = blockIdx.y;
    int b  = bh >> 1;
    int d  = (bh & 1) * 16 + wave * 2 + (lane >> 4);
    int n  = lane & 15;

    float a = -__expf(A_log[d * DSTATE + n]);
    size_t base = (size_t)b * SEQ + (size_t)c * CHUNK;
    const float* dtp = dt + base * DINNER + d;
    const float* xcp = xc + base * DINNER + d;
    const float* Bp  = Bm + base * DSTATE + n;
    const float* Cp  = Cm + base * DSTATE + n;
    float*       yp  = ybuf + base * DINNER + d;

    size_t si = ((((size_t)b * DINNER) + d) * DSTATE + n) * NCHUNK + c;
    float h = hinit[si];

    for (int l = 0; l < CHUNK; ++l) {
        float dtv = dtp[(size_t)l * DINNER];
        float xv  = xcp[(size_t)l * DINNER];
        float bv  = Bp[(size_t)l * DSTATE];
        float cv  = Cp[(size_t)l * DSTATE];
        float dA  = __expf(dtv * a);
        h = fmaf(dA, h, dtv * bv * xv);
        float contrib = h * cv;
        contrib += __shfl_xor(contrib, 1, 32);
        contrib += __shfl_xor(contrib, 2, 32);
        contrib += __shfl_xor(contrib, 4, 32);
        contrib += __shfl_xor(contrib, 8, 32);
        if (n == 0) yp[(size_t)l * DINNER] = contrib;
    }
}

// ---------------------------------------------------------------------------
// Kernel 4: y = (y + x*D) * silu(z); out_proj; + residual; final layernorm
// ---------------------------------------------------------------------------
__global__ __launch_bounds__(256) void post_kernel(
    const float* __restrict__ ybuf, const float* __restrict__ xc,
    const float* __restrict__ zbuf, const float* __restrict__ res,
    const float* __restrict__ Dp, const float* __restrict__ outw,
    const float* __restrict__ nfg, const float* __restrict__ nfb,
    float* __restrict__ hf)
{
    int t = blockIdx.x * 256 + threadIdx.x;
    if (t >= NTOK) return;

    float y2[DINNER];
    for (int d = 0; d < DINNER; ++d) {
        float yv = ybuf[(size_t)t * DINNER + d] + xc[(size_t)t * DINNER + d] * Dp[d];
        y2[d] = yv * silu_f(zbuf[(size_t)t * DINNER + d]);
    }

    float o[16];
    for (int j = 0; j < 16; ++j) {
        float s = 0.0f;
#pragma unroll
        for (int d = 0; d < DINNER; ++d) s += outw[j * DINNER + d] * y2[d];
        o[j] = s + res[(size_t)t * 16 + j];
    }

    float hv[16];
    layernorm16(o, nfg, nfb, hv);
#pragma unroll
    for (int j = 0; j < 16; ++j) hf[(size_t)t * 16 + j] = hv[j];
}

// ---------------------------------------------------------------------------
// Kernel 5: per (batch, p-tile) f32 WMMA GEMM over K = 2048 with FOUR
// independent accumulators to break the WMMA D->C dependency chain.
// Epilogue via LDS fuses +lin_b and the 16->8 head projection + head_b.
// ---------------------------------------------------------------------------
__global__ __launch_bounds__(32) void head_gemm_kernel(
    const float* __restrict__ hf, const float* __restrict__ linw,
    const float* __restrict__ linb, const float* __restrict__ headw,
    const float* __restrict__ headb, float* __restrict__ out)
{
    __shared__ float tile[16][17];

    int lane = threadIdx.x;
    int pt   = blockIdx.x;      // 0..44
    int b    = blockIdx.y;      // 0..127
    int m    = lane & 15;
    int half = lane >> 4;
    int kb   = half * 2;
    int p0   = pt * 16;

    const float* wrow = linw + (size_t)(p0 + m) * SEQ + kb;
    const float* hcol = hf + ((size_t)b * SEQ + kb) * DMODEL + m;

    v8f c0 = {0.f,0.f,0.f,0.f,0.f,0.f,0.f,0.f};
    v8f c1 = c0, c2 = c0, c3 = c0;

    for (int kc = 0; kc < SEQ; kc += 64) {
        __builtin_prefetch(wrow + kc + 64, 0, 0);   // global_prefetch_b8
#pragma unroll
        for (int k0 = 0; k0 < 64; k0 += 16) {
            v2f a0, a1, a2, a3, b0, b1, b2, b3;
            a0.x = wrow[kc + k0 +  0]; a0.y = wrow[kc + k0 +  1];
            a1.x = wrow[kc + k0 +  4]; a1.y = wrow[kc + k0 +  5];
            a2.x = wrow[kc + k0 +  8]; a2.y = wrow[kc + k0 +  9];
            a3.x = wrow[kc + k0 + 12]; a3.y = wrow[kc + k0 + 13];
            b0.x = hcol[(size_t)(kc + k0 +  0) * DMODEL];
            b0.y = hcol[(size_t)(kc + k0 +  1) * DMODEL];
            b1.x = hcol[(size_t)(kc + k0 +  4) * DMODEL];
            b1.y = hcol[(size_t)(kc + k0 +  5) * DMODEL];
            b2.x = hcol[(size_t)(kc + k0 +  8) * DMODEL];
            b2.y = hcol[(size_t)(kc + k0 +  9) * DMODEL];
            b3.x = hcol[(size_t)(kc + k0 + 12) * DMODEL];
            b3.y = hcol[(size_t)(kc + k0 + 13) * DMODEL];
            c0 = __builtin_amdgcn_wmma_f32_16x16x4_f32(false, a0, false, b0, (short)0, c0, false, false);
            c1 = __builtin_amdgcn_wmma_f32_16x16x4_f32(false, a1, false, b1, (short)0, c1, false, false);
            c2 = __builtin_amdgcn_wmma_f32_16x16x4_f32(false, a2, false, b2, (short)0, c2, false, false);
            c3 = __builtin_amdgcn_wmma_f32_16x16x4_f32(false, a3, false, b3, (short)0, c3, false, false);
        }
    }

    v8f c = (c0 + c1) + (c2 + c3);

    // D layout: VGPR j -> (M = j + half*8, N = m)
#pragma unroll
    for (int j = 0; j < 8; ++j) tile[j + half * 8][m] = c[j];
    __syncthreads();

    for (int pe = lane; pe < 128; pe += 32) {
        int p = pe >> 3;
        int e = pe & 7;
        float lb = linb[p0 + p];
        float s = headb[e];
#pragma unroll
        for (int d2 = 0; d2 < 16; ++d2)
            s += (tile[p][d2] + lb) * headw[e * 16 + d2];
        out[((size_t)b * PRED + (p0 + p)) * ENC + e] = s;
    }
}

// ---------------------------------------------------------------------------
extern "C" void kernel_launch(void* const* d_in, const int* in_sizes, int n_in,
                              void* d_out, int out_size, void* d_ws, size_t ws_size,
                              hipStream_t stream) {
    const float* x       = (const float*)d_in[0];
    const float* proj_w1 = (const float*)d_in[1];
    const float* proj_b1 = (const float*)d_in[2];
    const float* proj_w2 = (const float*)d_in[3];
    const float* proj_b2 = (const float*)d_in[4];
    const float* ln_g    = (const float*)d_in[5];
    const float* ln_b    = (const float*)d_in[6];
    const float* bn_g    = (const float*)d_in[7];
    const float* bn_b    = (const float*)d_in[8];
    const float* in_w    = (const float*)d_in[9];
    const float* conv_w  = (const float*)d_in[10];
    const float* conv_b  = (const float*)d_in[11];
    const float* xproj_w = (const float*)d_in[12];
    const float* dt_w    = (const float*)d_in[13];
    const float* dt_b    = (const float*)d_in[14];
    const float* A_log   = (const float*)d_in[15];
    const float* Dp      = (const float*)d_in[16];
    const float* out_w   = (const float*)d_in[17];
    const float* nf_g    = (const float*)d_in[18];
    const float* nf_b    = (const float*)d_in[19];
    const float* lin_w   = (const float*)d_in[20];
    const float* lin_b   = (const float*)d_in[21];
    const float* head_w  = (const float*)d_in[22];
    const float* head_b  = (const float*)d_in[23];
    float* out = (float*)d_out;

    const size_t N  = (size_t)NTOK;
    const size_t NS = (size_t)BATCHN * DINNER * DSTATE * NCHUNK;  // 524288
    float* ws    = (float*)d_ws;
    float* res   = ws;               // N*16
    float* xpre  = res  + N * 16;    // N*32
    float* zbuf  = xpre + N * 32;    // N*32
    float* xc    = zbuf + N * 32;    // N*32
    float* dt    = xc   + N * 32;    // N*32
    float* Bm    = dt   + N * 32;    // N*16
    float* Cm    = Bm   + N * 16;    // N*16
    float* hend  = Cm   + N * 16;    // NS
    float* pprod = hend + NS;        // NS
    float* hinit = pprod + NS;       // NS
    float* ybuf  = xpre;             // alias: xpre dead after conv kernel
    float* hf    = dt;               // alias: dt dead after scan

    const int tok_blocks = NTOK / 256;

    front_kernel<<<tok_blocks, 256, 0, stream>>>(
        x, proj_w1, proj_b1, proj_w2, proj_b2, ln_g, ln_b, bn_g, bn_b, in_w,
        res, xpre, zbuf);

    conv_proj_kernel<<<tok_blocks, 256, 0, stream>>>(
        xpre, conv_w, conv_b, xproj_w, dt_w, dt_b, xc, dt, Bm, Cm);

    scan_local_kernel<<<dim3(NCHUNK, BATCHN * 2), 256, 0, stream>>>(
        dt, xc, Bm, A_log, hend, pprod);

    scan_combine_kernel<<<(BATCHN * DINNER * DSTATE) / 256, 256, 0, stream>>>(
        hend, pprod, hinit);

    scan_final_kernel<<<dim3(NCHUNK, BATCHN * 2), 256, 0, stream>>>(
        dt, xc, Bm, Cm, A_log, hinit, ybuf);

    post_kernel<<<tok_blocks, 256, 0, stream>>>(
        ybuf, xc, zbuf, res, Dp, out_w, nf_g, nf_b, hf);

    head_gemm_kernel<<<dim3(PRED / 16, BATCHN), 32, 0, stream>>>(
        hf, lin_w, lin_b, head_w, head_b, out);
}